// DNADeBruijnClassifier_73632919323008
// MI455X (gfx1250) — compile-verified
//
#include <hip/hip_runtime.h>
#include <hip/hip_bf16.h>

typedef __attribute__((ext_vector_type(2))) float v2f;
typedef __attribute__((ext_vector_type(8))) float v8f;
typedef __attribute__((ext_vector_type(4))) unsigned int u32x4;
typedef __attribute__((ext_vector_type(8))) int i32x8;
typedef __attribute__((ext_vector_type(4))) int i32x4;

constexpr int Bg  = 16;        // graphs (batch)
constexpr int Nn  = 1024;      // nodes per graph
constexpr int Ee  = 8192;      // edges per graph
constexpr int ENe = Ee + Nn;   // edges + self loops
constexpr int Hh  = 4;         // attention heads
constexpr int Ch  = 256;       // channels per head
constexpr int Dd  = 1024;      // Hh*Ch
constexpr int BN  = Bg * Nn;   // total node rows = 16384

// GEMM macro-tile
constexpr int BM = 128;        // rows per block
constexpr int BNT = 64;        // cols per block
constexpr int KC = 32;         // K chunk (all K in this net are multiples of 32)
constexpr int LDA_S = KC + 1;  // LDS row stride for A (TDM pad: 32 DW + 1 DW)
constexpr int LDB_S = BNT + 1; // LDS row stride for B (TDM pad: 64 DW + 1 DW)

__device__ __forceinline__ v8f wmma4(v2f a, v2f b, v8f c) {
  return __builtin_amdgcn_wmma_f32_16x16x4_f32(false, a, false, b,
                                               (short)0, c, false, false);
}

// ---------------------------------------------------------------------------
// TDM 2D tile load: global (strided rows) -> LDS (rows padded +1 DWORD).
// D# packing per CDNA5 ISA. 2-D tile: tile_dim2=0,
// trailing descriptor groups zero. OOB rows/cols (beyond tensor_dim) are
// zero-filled by the TDM.
// padIv: pad_interval code (4 -> every 32 DWORDs, 5 -> every 64 DWORDs),
// pad_amount=0 -> +1 DWORD, giving LDS row strides 33 / 65.
// This toolchain declares the 6-arg builtin:
//   (u32x4 g0, i32x8 g1, i32x4, i32x4, i32x8, i32 cpol)
// ---------------------------------------------------------------------------
__device__ __forceinline__ void tdm_load2d(unsigned ldsOff, const float* gptr,
                                           unsigned tens0, unsigned tens1,
                                           unsigned tile0, unsigned tile1,
                                           unsigned stride0, unsigned padIv)
{
  unsigned long long ga = (unsigned long long)(uintptr_t)gptr;
  u32x4 g0;
  g0.x = 1u;                                            // count=1, user mode
  g0.y = ldsOff;                                        // lds_addr (bytes)
  g0.z = (unsigned)ga;                                  // global_addr[31:0]
  g0.w = ((unsigned)(ga >> 32) & 0x01FFFFFFu) | (2u << 30); // [56:32] | type=2
  i32x8 g1;
  g1[0] = (int)((2u << 16) | (1u << 20) | (padIv << 22));   // dsz=4B, pad_en
  g1[1] = (int)((tens0 & 0xFFFFu) << 16);               // tensor_dim0[15:0]
  g1[2] = (int)((tens0 >> 16) | ((tens1 & 0xFFFFu) << 16));
  g1[3] = (int)((tens1 >> 16) | (tile0 << 16));         // tile_dim0
  g1[4] = (int)(tile1 & 0xFFFFu);                       // tile_dim1, tile_dim2=0
  g1[5] = (int)stride0;                                 // tensor_dim0_stride
  g1[6] = 0;
  g1[7] = 0;
  i32x4 z4 = {0, 0, 0, 0};
  i32x8 z8 = {0, 0, 0, 0, 0, 0, 0, 0};
  __builtin_amdgcn_tensor_load_to_lds(g0, g1, z4, z4, z8, 0);
}

// ---------------------------------------------------------------------------
// fp32 WMMA GEMM: C[M,N] = A[M,K] @ W[K,N] (+bias) (relu?)
// Block: 8 waves, 128x64 macro-tile, K chunks of 32 staged in LDS by the
// Tensor Data Mover (double-buffered; DMA of chunk i+1 overlaps compute of
// chunk i). Each wave owns a 32x32 sub-tile = 2x2 WMMA fragments.
// Requires K % 32 == 0 (true for all GEMMs here). M,N ragged edges handled by
// TDM zero-fill on loads and guards on stores.
// ---------------------------------------------------------------------------
__global__ void __launch_bounds__(256)
gemm_wmma_f32(const float* __restrict__ A, int lda,
              const float* __restrict__ W,   // [K,N] row-major
              const float* __restrict__ bias,
              float* __restrict__ C, int ldc,
              int M, int N, int K, int relu)
{
  __shared__ float As[2][BM * LDA_S];   // 2 x 16.5 KB
  __shared__ float Bs[2][KC * LDB_S];   // 2 x  8.1 KB

  const int ntN  = (N + BNT - 1) / BNT;
  const int bm   = blockIdx.x / ntN;
  const int bn   = blockIdx.x - bm * ntN;
  const int row0 = bm * BM, col0 = bn * BNT;

  const int tid  = threadIdx.x;
  const int lane = tid & 31;
  const int wv   = tid >> 5;
  const int wm   = wv >> 1;            // 0..3 : 32-row band
  const int wn   = wv & 1;             // 0..1 : 32-col band
  const int l16  = lane & 15;
  const int hi   = lane >> 4;          // 0: k+0/1 lanes, 1: k+2/3 lanes

  auto issue_tdm = [&](int kk, int buf) {
    tdm_load2d((unsigned)(uintptr_t)&As[buf][0],
               A + (size_t)row0 * lda + kk,
               (unsigned)(K - kk), (unsigned)(M - row0),
               KC, BM, (unsigned)lda, 4);
    tdm_load2d((unsigned)(uintptr_t)&Bs[buf][0],
               W + (size_t)kk * N + col0,
               (unsigned)(N - col0), (unsigned)(K - kk),
               BNT, KC, (unsigned)N, 5);
  };

  const int nch = K / KC;
  if (tid < 32) {                       // wave 0 drives the TDM
    issue_tdm(0, 0);
    __builtin_amdgcn_s_wait_tensorcnt(0);
  }
  __syncthreads();

  v8f acc00 = {0.f,0.f,0.f,0.f,0.f,0.f,0.f,0.f};
  v8f acc01 = acc00, acc10 = acc00, acc11 = acc00;

  const int ar0 = (wm << 5) + l16;      // A fragment rows
  const int ar1 = ar0 + 16;
  const int bc0 = (wn << 5) + l16;      // B fragment cols
  const int bc1 = bc0 + 16;

  for (int ch = 0; ch < nch; ++ch) {
    const int buf = ch & 1;
    if ((ch + 1 < nch) && tid < 32)     // prefetch next chunk via TDM
      issue_tdm((ch + 1) * KC, buf ^ 1);

    const float* As_ = &As[buf][0];
    const float* Bs_ = &Bs[buf][0];
    #pragma unroll
    for (int k = 0; k < KC; k += 4) {
      const int ka = k + (hi << 1);
      v2f a0, a1, b0, b1;
      a0.x = As_[ar0 * LDA_S + ka];
      a0.y = As_[ar0 * LDA_S + ka + 1];
      a1.x = As_[ar1 * LDA_S + ka];
      a1.y = As_[ar1 * LDA_S + ka + 1];
      b0.x = Bs_[ka * LDB_S + bc0];
      b0.y = Bs_[(ka + 1) * LDB_S + bc0];
      b1.x = Bs_[ka * LDB_S + bc1];
      b1.y = Bs_[(ka + 1) * LDB_S + bc1];
      acc00 = wmma4(a0, b0, acc00);
      acc01 = wmma4(a0, b1, acc01);
      acc10 = wmma4(a1, b0, acc10);
      acc11 = wmma4(a1, b1, acc11);
    }
    __syncthreads();                    // all waves done with buf
    if (ch + 1 < nch) {
      if (tid < 32) __builtin_amdgcn_s_wait_tensorcnt(0);
      __syncthreads();                  // next buffer is ready for everyone
    }
  }

  auto store_tile = [&](const v8f& acc, int tr0, int tc0) {
    const int col = col0 + tc0 + l16;
    if (col >= N) return;
    const float bv = bias ? bias[col] : 0.f;
    const int r0 = row0 + tr0 + (hi << 3);
    #pragma unroll
    for (int r = 0; r < 8; ++r) {
      const int rr = r0 + r;
      if (rr < M) {
        float v = acc[r] + bv;
        if (relu) v = fmaxf(v, 0.f);
        C[(size_t)rr * ldc + col] = v;
      }
    }
  };
  store_tile(acc00, wm << 5, wn << 5);
  store_tile(acc01, wm << 5, (wn << 5) + 16);
  store_tile(acc10, (wm << 5) + 16, wn << 5);
  store_tile(acc11, (wm << 5) + 16, (wn << 5) + 16);
}

// --------------------------- small utility kernels -------------------------
__global__ void fill_f32(float* p, float v, int n) {
  int i = blockIdx.x * blockDim.x + threadIdx.x;
  if (i < n) p[i] = v;
}
__global__ void fill_u32(unsigned* p, unsigned v, int n) {
  int i = blockIdx.x * blockDim.x + threadIdx.x;
  if (i < n) p[i] = v;
}
// init AGG rows with [fwd_bias | bwd_bias] so scatter-adds land on bias
__global__ void agg_init(float* __restrict__ agg,
                         const float* __restrict__ fb,
                         const float* __restrict__ bb)
{
  long i = (long)blockIdx.x * blockDim.x + threadIdx.x;   // over BN*2*Dd
  int col = (int)(i & (2 * Dd - 1));
  agg[i] = (col < Dd) ? fb[col] : bb[col - Dd];
}

// a[n,h] = sum_c h[n, h*Ch+c] * att[h,c] ; one wave per (node,head)
__global__ void attn_scores(const float* __restrict__ hbuf,
                            const float* __restrict__ att,   // [Hh,Ch]
                            float* __restrict__ out)         // [BN,Hh]
{
  const long wid = (((long)blockIdx.x * blockDim.x + threadIdx.x) >> 5);
  const int lane = threadIdx.x & 31;
  if (wid >= (long)BN * Hh) return;
  const int m = (int)(wid >> 2), h = (int)(wid & 3);
  const float* hp = hbuf + (size_t)m * Dd + h * Ch;
  const float* ap = att + h * Ch;
  float s = 0.f;
  for (int i = lane; i < Ch; i += 32) s += hp[i] * ap[i];
  #pragma unroll
  for (int off = 16; off > 0; off >>= 1) s += __shfl_xor(s, off, 32);
  if (lane == 0) out[wid] = s;
}

__device__ __forceinline__ void edge_st(const int* __restrict__ ei,
                                        int b, int e, int& s, int& t)
{
  if (e < Ee) {
    const int* p = ei + (size_t)b * 2 * Ee;
    s = p[e]; t = p[Ee + e];
  } else {
    s = t = e - Ee;                                // self loop
  }
}

// alpha = leaky_relu(a_src[s]+a_dst[t]); segment-max via ordered-uint atomicMax
__global__ void edge_alpha_max(const float* __restrict__ asrc,
                               const float* __restrict__ adst,
                               const int* __restrict__ ei,
                               float* __restrict__ alpha,
                               unsigned* __restrict__ amax)
{
  const long idx = (long)blockIdx.x * blockDim.x + threadIdx.x;
  if (idx >= (long)Bg * ENe * Hh) return;
  const int h = (int)(idx & 3);
  const int e = (int)((idx >> 2) % ENe);
  const int b = (int)(idx / ((long)ENe * Hh));
  int s, t; edge_st(ei, b, e, s, t);
  float a = asrc[(b * Nn + s) * Hh + h] + adst[(b * Nn + t) * Hh + h];
  a = (a > 0.f) ? a : 0.2f * a;                    // leaky slope 0.2
  alpha[idx] = a;
  unsigned u = __float_as_uint(a);
  u = (u & 0x80000000u) ? ~u : (u | 0x80000000u);  // order-preserving map
  atomicMax(&amax[(b * Nn + t) * Hh + h], u);
}

// alpha <- exp(alpha - max); segment-sum via atomicAdd f32
__global__ void edge_exp_sum(float* __restrict__ alpha,
                             const unsigned* __restrict__ amax,
                             float* __restrict__ denom,
                             const int* __restrict__ ei)
{
  const long idx = (long)blockIdx.x * blockDim.x + threadIdx.x;
  if (idx >= (long)Bg * ENe * Hh) return;
  const int h = (int)(idx & 3);
  const int e = (int)((idx >> 2) % ENe);
  const int b = (int)(idx / ((long)ENe * Hh));
  int s, t; edge_st(ei, b, e, s, t); (void)s;
  unsigned u = amax[(b * Nn + t) * Hh + h];
  float mx = (u & 0x80000000u) ? __uint_as_float(u ^ 0x80000000u)
                               : __uint_as_float(~u);
  float v = __expf(alpha[idx] - mx);
  alpha[idx] = v;
  atomicAdd(&denom[(b * Nn + t) * Hh + h], v);
}

// out[t] += h[s] * alpha/(denom+eps) ; one wave per (edge,head), 256 channels
__global__ void edge_scatter(const float* __restrict__ hsrc,
                             const float* __restrict__ alpha,
                             const float* __restrict__ denom,
                             const int* __restrict__ ei,
                             float* __restrict__ agg, int colOff)
{
  const long wid = (((long)blockIdx.x * blockDim.x + threadIdx.x) >> 5);
  const int lane = threadIdx.x & 31;
  if (wid >= (long)Bg * ENe * Hh) return;
  const int h = (int)(wid & 3);
  const int e = (int)((wid >> 2) % ENe);
  const int b = (int)(wid / ((long)ENe * Hh));
  int s, t; edge_st(ei, b, e, s, t);
  const float a = alpha[wid] / (denom[(b * Nn + t) * Hh + h] + 1e-16f);
  const float* sp = hsrc + (size_t)(b * Nn + s) * Dd + h * Ch;
  float* dp = agg + (size_t)(b * Nn + t) * (2 * Dd) + colOff + h * Ch;
  for (int i = lane; i < Ch; i += 32)
    atomicAdd(&dp[i], sp[i] * a);
}

// training-mode BN stats per (graph, channel); coalesced column walk
__global__ void colstats(const float* __restrict__ X,
                         float* __restrict__ mu, float* __restrict__ var)
{
  const int b = blockIdx.x >> 2;
  const int c = ((blockIdx.x & 3) << 8) + threadIdx.x;
  const float* p = X + (size_t)b * Nn * Dd + c;
  float s = 0.f, s2 = 0.f;
  for (int n = 0; n < Nn; ++n) {
    float v = p[(size_t)n * Dd];
    s += v; s2 += v * v;
  }
  float m = s * (1.f / Nn);
  mu[b * Dd + c]  = m;
  var[b * Dd + c] = s2 * (1.f / Nn) - m * m;     // biased var (ddof=0)
}

// x_res <- gamma*(h-mu)*rsqrt(var+eps)+beta + x_res
__global__ void bn_apply_res(const float* __restrict__ h,
                             const float* __restrict__ mu,
                             const float* __restrict__ var,
                             const float* __restrict__ g,
                             const float* __restrict__ be,
                             float* __restrict__ xres)
{
  long i = (long)blockIdx.x * blockDim.x + threadIdx.x;   // over BN*Dd
  if (i >= (long)BN * Dd) return;
  const int c = (int)(i & (Dd - 1));
  const int b = (int)((i >> 10) >> 10);
  float v = h[i];
  float y = g[c] * (v - mu[b * Dd + c]) * rsqrtf(var[b * Dd + c] + 1e-5f) + be[c];
  xres[i] = y + xres[i];
}

__global__ void pool_mean(const float* __restrict__ X, float* __restrict__ out)
{
  const int b = blockIdx.x >> 2;
  const int c = ((blockIdx.x & 3) << 8) + threadIdx.x;
  const float* p = X + (size_t)b * Nn * Dd + c;
  float s = 0.f;
  for (int n = 0; n < Nn; ++n) s += p[(size_t)n * Dd];
  out[b * Dd + c] = s * (1.f / Nn);
}

// ---------------------------------------------------------------------------
extern "C" void kernel_launch(void* const* d_in, const int* in_sizes, int n_in,
                              void* d_out, int out_size, void* d_ws, size_t ws_size,
                              hipStream_t stream)
{
  (void)in_sizes; (void)n_in; (void)out_size; (void)ws_size;
  const float* x_in   = (const float*)d_in[0];
  const int*   fwd_ei = (const int*)d_in[1];
  const int*   bwd_ei = (const int*)d_in[2];
  auto F = [&](int i) { return (const float*)d_in[i]; };

  // params flattened in jax pytree (sorted-key) order:
  // layer dict keys: bn_b, bn_g, bwd_W, bwd_ad, bwd_as, bwd_b,
  //                  fwd_W, fwd_ad, fwd_as, fwd_b, merge_W, merge_b, [res_W, res_b]
  struct LayerP { int bn_b, bn_g, bwd_W, bwd_ad, bwd_as, bwd_b,
                      fwd_W, fwd_ad, fwd_as, fwd_b, merge_W, merge_b, res_W, res_b; };
  const LayerP L1 = { 4, 5, 6, 7, 8, 9,10,11,12,13,14,15,16,17};
  const LayerP L2 = {18,19,20,21,22,23,24,25,26,27,28,29,-1,-1};
  const int MLPW1 = 30, MLPB1 = 31, MLPW2 = 32, MLPB2 = 33, MLPW3 = 34, MLPB3 = 35;

  // ----- workspace layout (f32 elements) -----
  float* base = (float*)d_ws;
  size_t off = 0;
  auto alloc = [&](size_t n) { float* p = base + off; off += n; return p; };
  float*    HF    = alloc((size_t)BN * Dd);
  float*    HB    = alloc((size_t)BN * Dd);
  float*    AGG   = alloc((size_t)BN * 2 * Dd);
  float*    HOUT  = alloc((size_t)BN * Dd);
  float*    X1    = alloc((size_t)BN * Dd);        // residual carrier / layer output
  float*    ASRC  = alloc((size_t)BN * Hh);
  float*    ADST  = alloc((size_t)BN * Hh);
  unsigned* AMAX  = (unsigned*)alloc((size_t)BN * Hh);
  float*    DENOM = alloc((size_t)BN * Hh);
  float*    ALPHA = alloc((size_t)Bg * ENe * Hh);
  float*    MU    = alloc((size_t)Bg * Dd);
  float*    VAR   = alloc((size_t)Bg * Dd);
  float*    POOL  = alloc((size_t)Bg * Dd);
  float*    M1    = alloc((size_t)Bg * 512);
  float*    M2    = alloc((size_t)Bg * 256);

  const dim3 blk(256);
  auto cdivl = [](long a, long b) { return (a + b - 1) / b; };

  auto gemm = [&](const float* Aa, int lda, const float* Wm, const float* bias,
                  float* Cc, int ldc, int M, int N, int K, int relu) {
    long blocks = cdivl(M, BM) * cdivl(N, BNT);
    gemm_wmma_f32<<<dim3((unsigned)blocks), blk, 0, stream>>>(
        Aa, lda, Wm, bias, Cc, ldc, M, N, K, relu);
  };

  const long nEdgeTh = (long)Bg * ENe * Hh;        // 589,824
  const long nAttnW  = (long)BN * Hh;              //  65,536 waves

  auto layer = [&](const float* xin, int K, const LayerP& P) {
    // node transforms h = x @ W (no bias here; GAT bias added via AGG init)
    gemm(xin, K, F(P.fwd_W), nullptr, HF, Dd, BN, Dd, K, 0);
    gemm(xin, K, F(P.bwd_W), nullptr, HB, Dd, BN, Dd, K, 0);
    // residual path into X1 (projection for layer 1; identity already in X1 otherwise)
    if (P.res_W >= 0)
      gemm(xin, K, F(P.res_W), F(P.res_b), X1, Dd, BN, Dd, K, 0);
    // aggregation target initialized to [fwd_b | bwd_b]
    agg_init<<<dim3((unsigned)cdivl((long)BN * 2 * Dd, 256)), blk, 0, stream>>>(
        AGG, F(P.fwd_b), F(P.bwd_b));

    auto gat_dir = [&](const float* hbuf, const float* as, const float* ad,
                       const int* ei, int colOff) {
      attn_scores<<<dim3((unsigned)cdivl(nAttnW * 32, 256)), blk, 0, stream>>>(hbuf, as, ASRC);
      attn_scores<<<dim3((unsigned)cdivl(nAttnW * 32, 256)), blk, 0, stream>>>(hbuf, ad, ADST);
      fill_u32<<<dim3((unsigned)cdivl(BN * Hh, 256)), blk, 0, stream>>>(AMAX, 0u, BN * Hh);
      fill_f32<<<dim3((unsigned)cdivl(BN * Hh, 256)), blk, 0, stream>>>(DENOM, 0.f, BN * Hh);
      edge_alpha_max<<<dim3((unsigned)cdivl(nEdgeTh, 256)), blk, 0, stream>>>(
          ASRC, ADST, ei, ALPHA, AMAX);
      edge_exp_sum<<<dim3((unsigned)cdivl(nEdgeTh, 256)), blk, 0, stream>>>(
          ALPHA, AMAX, DENOM, ei);
      edge_scatter<<<dim3((unsigned)cdivl(nEdgeTh * 32, 256)), blk, 0, stream>>>(
          hbuf, ALPHA, DENOM, ei, AGG, colOff);
    };
    gat_dir(HF, F(P.fwd_as), F(P.fwd_ad), fwd_ei, 0);
    gat_dir(HB, F(P.bwd_as), F(P.bwd_ad), bwd_ei, Dd);

    // merge + relu
    gemm(AGG, 2 * Dd, F(P.merge_W), F(P.merge_b), HOUT, Dd, BN, Dd, 2 * Dd, 1);
    // train-mode batchnorm + residual add (into X1)
    colstats<<<dim3(Bg * (Dd / 256)), blk, 0, stream>>>(HOUT, MU, VAR);
    bn_apply_res<<<dim3((unsigned)cdivl((long)BN * Dd, 256)), blk, 0, stream>>>(
        HOUT, MU, VAR, F(P.bn_g), F(P.bn_b), X1);
  };

  layer(x_in, 128, L1);   // X1 = layer1 output (res proj + bn(h))
  layer(X1,   Dd,  L2);   // X1 = layer2 output (identity residual, in place)

  // global mean pool + MLP head
  pool_mean<<<dim3(Bg * (Dd / 256)), blk, 0, stream>>>(X1, POOL);
  gemm(POOL, Dd,  F(MLPW1), F(MLPB1), M1, 512, Bg, 512, Dd,  1);
  gemm(M1,   512, F(MLPW2), F(MLPB2), M2, 256, Bg, 256, 512, 1);
  gemm(M2,   256, F(MLPW3), F(MLPB3), (float*)d_out, 1000, Bg, 1000, 256, 0);
}